// MambaBlock_86595130622298
// MI455X (gfx1250) — compile-verified
//
#include <hip/hip_runtime.h>
#include <math.h>

// ---------------- problem dims (compile-time) ----------------
#define D_MODEL 1024
#define D_STATE 16
#define D_INNER 2048
#define DT_RANK 64
#define D_CONV  4
#define BB      8
#define LL      2048
#define ROWS    (BB * LL)                 // 16384 token rows
#define DPROJ   (DT_RANK + 2 * D_STATE)   // 96

typedef __attribute__((ext_vector_type(2))) float v2f;
typedef __attribute__((ext_vector_type(8))) float v8f;

#if defined(__has_builtin)
#if __has_builtin(__builtin_amdgcn_global_load_async_to_lds_b32) && \
    __has_builtin(__builtin_amdgcn_s_wait_asynccnt)
#define USE_ASYNC_COPY 1
#endif
#endif
#ifndef USE_ASYNC_COPY
#define USE_ASYNC_COPY 0
#endif

typedef __attribute__((address_space(1))) int as1_i32;
typedef __attribute__((address_space(3))) int as3_i32;

__device__ __forceinline__ float sigmoidf_(float x) { return 1.f / (1.f + __expf(-x)); }

// ---------------- RMSNorm: one block per row of 1024 ----------------
__global__ void rmsnorm_kernel(const float* __restrict__ x,
                               const float* __restrict__ w,
                               float* __restrict__ xn) {
    const int row = blockIdx.x;
    const int t   = threadIdx.x;                       // 256 threads, 4 floats each
    const float4 v = ((const float4*)(x + (size_t)row * D_MODEL))[t];
    float ss = v.x * v.x + v.y * v.y + v.z * v.z + v.w * v.w;
#pragma unroll
    for (int o = 16; o > 0; o >>= 1) ss += __shfl_xor(ss, o, 32);
    __shared__ float sred[8];
    if ((t & 31) == 0) sred[t >> 5] = ss;
    __syncthreads();
    if (t == 0) {
        float tot = 0.f;
#pragma unroll
        for (int i = 0; i < 8; ++i) tot += sred[i];
        sred[0] = rsqrtf(tot / (float)D_MODEL + 1.1920928955078125e-07f);
    }
    __syncthreads();
    const float s = sred[0];
    const float4 wv = ((const float4*)w)[t];
    float4 o;
    o.x = v.x * s * wv.x;  o.y = v.y * s * wv.y;
    o.z = v.z * s * wv.z;  o.w = v.w * s * wv.w;
    ((float4*)(xn + (size_t)row * D_MODEL))[t] = o;
}

// ---------------- generic fp32 WMMA GEMM: C = A[MxK] * B[KxN] ----------------
// 256 threads = 8 waves; block tile 128(M) x 64(N); wave tile 16(M) x 64(N).
// B panel (32 x 64) staged in LDS PRE-SWIZZLED into WMMA fragment order:
//   fragment block (k4, nt), lane ln -> dwords (k4*4+nt)*96 + ln*2 + {0,1}
// so each B fragment is one conflict-free ds_load_b64 (no packing movs).
// The 96-dword block stride also makes the staging writes conflict-free.
// Staging uses gfx1250 async global->LDS copies (ASYNCcnt) when available.
// K must be a multiple of 32 (true here: 1024 / 2048 / 64).
// EPI: 0 = plain, 1 = softplus(acc + 2*bias[n]), 2 = acc + resid[m,n]
template <int EPI>
__global__ void wmma_gemm_f32(const float* __restrict__ A, const float* __restrict__ Bm,
                              float* __restrict__ C, int N, int K,
                              int lda, int ldb, int ldc,
                              const float* __restrict__ bias,
                              const float* __restrict__ resid) {
    __shared__ float sB[8 * 4 * 96];            // 12 KB, fragment-swizzled
    const int tid   = threadIdx.x;
    const int lane  = tid & 31;
    const int wave  = tid >> 5;
    const int n0    = blockIdx.x * 64;
    const int m0    = blockIdx.y * 128 + wave * 16;
    const int arow  = m0 + (lane & 15);
    const int khalf = (lane >> 4) << 1;         // 0 for lanes 0-15, 2 for 16-31
    const float* Arow = A + (size_t)arow * lda;

    // staging constants (per thread): r0 = K-row group, c = column in panel
    const int r0 = tid >> 6;                    // 0..3
    const int c  = tid & 63;
    const int jj = r0 & 1;
    const int hi = r0 >> 1;
    const int lt = (hi << 4) | (c & 15);        // destination lane of this element
    const int nt_s   = c >> 4;                  // destination N-subtile
    const int lbase  = nt_s * 96 + lt * 2 + jj; // dword, += 384 per k4
    const int gcol   = n0 + c;
    const bool inb   = (gcol < N);
    const bool fullN = (n0 + 64 <= N);

    v8f acc[4] = {};                            // 4 N-subtiles of 16x16 f32

    for (int kb = 0; kb < K; kb += 32) {
        __syncthreads();
        // ---- stage B panel (32 x 64) into swizzled LDS ----
#if USE_ASYNC_COPY
        if (fullN) {
#pragma unroll
            for (int i = 0; i < 8; ++i) {
                const size_t goff = (size_t)(kb + r0 + 4 * i) * ldb + gcol;
                __builtin_amdgcn_global_load_async_to_lds_b32(
                    (as1_i32*)(void*)(Bm + goff),
                    (as3_i32*)(void*)&sB[lbase + i * 384],
                    0, 0);
            }
            __builtin_amdgcn_s_wait_asynccnt(0);
        } else
#endif
        {
#pragma unroll
            for (int i = 0; i < 8; ++i) {
                const size_t goff = (size_t)(kb + r0 + 4 * i) * ldb + gcol;
                sB[lbase + i * 384] = inb ? Bm[goff] : 0.f;
            }
        }
        __syncthreads();

        // ---- preload all 8 A fragments of this chunk (issues 8 b64 loads) ----
        v2f av[8];
#pragma unroll
        for (int k4 = 0; k4 < 8; ++k4)
            av[k4] = *(const v2f*)(Arow + kb + k4 * 4 + khalf);

        // ---- 32 WMMAs: one conflict-free ds_load_b64 per B fragment ----
#pragma unroll
        for (int k4 = 0; k4 < 8; ++k4) {
#pragma unroll
            for (int nt = 0; nt < 4; ++nt) {
                const v2f bv = *(const v2f*)&sB[(k4 * 4 + nt) * 96 + (lane << 1)];
                acc[nt] = __builtin_amdgcn_wmma_f32_16x16x4_f32(
                    false, av[k4], false, bv, (short)0, acc[nt], false, false);
            }
        }
    }

    // C/D layout: VGPR v -> row m0+v (lanes 0-15) / m0+v+8 (lanes 16-31)
    const int rbase = m0 + ((lane >> 4) << 3);
    const int cbase = lane & 15;
#pragma unroll
    for (int nt = 0; nt < 4; ++nt) {
        const int col = n0 + nt * 16 + cbase;
        if (col < N) {
#pragma unroll
            for (int v = 0; v < 8; ++v) {
                const int row = rbase + v;
                float val = acc[nt][v];
                if (EPI == 1) {                 // delta = softplus(dt + b_dt); dt already has +b_dt
                    const float z = val + 2.f * bias[col];
                    val = (z > 20.f) ? z : log1pf(__expf(z));
                } else if (EPI == 2) {
                    val += resid[(size_t)row * ldc + col];
                }
                C[(size_t)row * ldc + col] = val;
            }
        }
    }
}

// ---------------- causal depthwise conv1d (k=4) + SiLU ----------------
// reads x_ssm = xp[:, 0:2048] (row stride 4096), writes x_act [rows, 2048]
__global__ void conv_silu_kernel(const float* __restrict__ xp,
                                 const float* __restrict__ cw,
                                 const float* __restrict__ cb,
                                 float* __restrict__ xact) {
    const size_t i  = (size_t)blockIdx.x * 256 + threadIdx.x;  // over ROWS*D_INNER
    const int    d  = (int)(i & (D_INNER - 1));
    const size_t rl = i >> 11;                                  // b*L + l
    const int    l  = (int)(rl & (LL - 1));
    float acc = cb[d];
    const float w0 = cw[d * 4 + 0], w1 = cw[d * 4 + 1];
    const float w2 = cw[d * 4 + 2], w3 = cw[d * 4 + 3];
    if (l >= 3) acc = fmaf(w0, xp[(rl - 3) * (2 * D_INNER) + d], acc);
    if (l >= 2) acc = fmaf(w1, xp[(rl - 2) * (2 * D_INNER) + d], acc);
    if (l >= 1) acc = fmaf(w2, xp[(rl - 1) * (2 * D_INNER) + d], acc);
    acc = fmaf(w3, xp[rl * (2 * D_INNER) + d], acc);
    xact[i] = acc * sigmoidf_(acc);
}

// ---------------- selective scan, fused gate + D term ----------------
// one lane per (b, d) channel; 16 SSM states in VGPRs; B/C vectors staged
// through LDS in 32-step chunks (broadcast reads).  Writes in-place over u.
__global__ void scan_kernel(const float* __restrict__ delta,
                            const float* __restrict__ xdbl,  // [rows, 96]: dt|B|C
                            const float* __restrict__ xp,    // gate = cols 2048..4095
                            const float* __restrict__ Alog,
                            const float* __restrict__ Dv,
                            float* __restrict__ uio) {       // in: u (x_act), out: y*silu(g)
    const int d = blockIdx.x * 256 + threadIdx.x;
    const int b = blockIdx.y;
    float a[D_STATE], h[D_STATE];
#pragma unroll
    for (int n = 0; n < D_STATE; ++n) {
        a[n] = -__expf(Alog[d * D_STATE + n]);
        h[n] = 0.f;
    }
    const float Dd = Dv[d];
    __shared__ float sB[32 * D_STATE];
    __shared__ float sC[32 * D_STATE];
    const size_t rowbase = (size_t)b * LL;

    for (int l0 = 0; l0 < LL; l0 += 32) {
        __syncthreads();
        for (int i = threadIdx.x; i < 32 * D_STATE; i += 256) {
            const int    s    = i >> 4, n = i & 15;
            const size_t base = (rowbase + l0 + s) * DPROJ;
            sB[i] = xdbl[base + DT_RANK + n];
            sC[i] = xdbl[base + DT_RANK + D_STATE + n];
        }
        __syncthreads();
#pragma unroll 4
        for (int s = 0; s < 32; ++s) {
            const size_t idx = (rowbase + l0 + s) * D_INNER + d;
            const float dt  = delta[idx];
            const float uu  = uio[idx];
            const float g   = xp[(rowbase + l0 + s) * (2 * D_INNER) + D_INNER + d];
            const float dtu = dt * uu;
            float y = 0.f;
#pragma unroll
            for (int n = 0; n < D_STATE; ++n) {
                const float dA = __expf(dt * a[n]);
                h[n] = fmaf(dA, h[n], dtu * sB[s * D_STATE + n]);
                y    = fmaf(h[n], sC[s * D_STATE + n], y);
            }
            const float yd = fmaf(uu, Dd, y);
            uio[idx] = yd * (g * sigmoidf_(g));
        }
    }
}

// ---------------- launcher ----------------
extern "C" void kernel_launch(void* const* d_in, const int* in_sizes, int n_in,
                              void* d_out, int out_size, void* d_ws, size_t ws_size,
                              hipStream_t stream) {
    (void)in_sizes; (void)n_in; (void)out_size; (void)ws_size;
    const float* x      = (const float*)d_in[0];
    const float* norm_w = (const float*)d_in[1];
    const float* w_in   = (const float*)d_in[2];
    const float* conv_w = (const float*)d_in[3];
    const float* conv_b = (const float*)d_in[4];
    const float* w_x    = (const float*)d_in[5];
    const float* w_dt   = (const float*)d_in[6];
    const float* b_dt   = (const float*)d_in[7];
    const float* A_log  = (const float*)d_in[8];
    const float* Dv     = (const float*)d_in[9];
    const float* w_out  = (const float*)d_in[10];
    float* out = (float*)d_out;

    char* ws = (char*)d_ws;
    const size_t sz_xn   = (size_t)ROWS * D_MODEL * 4;        //  64 MB
    const size_t sz_xp   = (size_t)ROWS * 2 * D_INNER * 4;    // 256 MB
    const size_t sz_xact = (size_t)ROWS * D_INNER * 4;        // 128 MB
    const size_t sz_xdbl = (size_t)ROWS * DPROJ * 4;          //   6 MB
    float* xn    = (float*)(ws);
    float* xp    = (float*)(ws + sz_xn);
    float* xact  = (float*)(ws + sz_xn + sz_xp);              // also x_fused (in-place)
    float* xdbl  = (float*)(ws + sz_xn + sz_xp + sz_xact);
    float* delta = (float*)(ws + sz_xn + sz_xp + sz_xact + sz_xdbl);

    // 1) RMSNorm
    rmsnorm_kernel<<<ROWS, 256, 0, stream>>>(x, norm_w, xn);

    // 2) in_proj: [16384,1024] @ [1024,4096] -> xp
    wmma_gemm_f32<0><<<dim3(2 * D_INNER / 64, ROWS / 128), 256, 0, stream>>>(
        xn, w_in, xp, 2 * D_INNER, D_MODEL, D_MODEL, 2 * D_INNER, 2 * D_INNER,
        nullptr, nullptr);

    // 3) causal depthwise conv + SiLU -> xact
    conv_silu_kernel<<<(ROWS * (size_t)D_INNER) / 256, 256, 0, stream>>>(
        xp, conv_w, conv_b, xact);

    // 4) x_proj: [16384,2048] @ [2048,96] -> xdbl (dt_low | B_ssm | C_ssm)
    wmma_gemm_f32<0><<<dim3(2, ROWS / 128), 256, 0, stream>>>(
        xact, w_x, xdbl, DPROJ, D_INNER, D_INNER, DPROJ, DPROJ,
        nullptr, nullptr);

    // 5) dt_proj + softplus(dt + b_dt) fused: [16384,64] @ [64,2048] -> delta
    wmma_gemm_f32<1><<<dim3(D_INNER / 64, ROWS / 128), 256, 0, stream>>>(
        xdbl, w_dt, delta, D_INNER, DT_RANK, DPROJ, D_INNER, D_INNER,
        b_dt, nullptr);

    // 6) selective scan + u*D + SiLU-gate, in-place over xact
    scan_kernel<<<dim3(D_INNER / 256, BB), 256, 0, stream>>>(
        delta, xdbl, xp, A_log, Dv, xact);

    // 7) out_proj + residual: [16384,2048] @ [2048,1024] + x -> out
    wmma_gemm_f32<2><<<dim3(D_MODEL / 64, ROWS / 128), 256, 0, stream>>>(
        xact, w_out, out, D_MODEL, D_INNER, D_INNER, D_MODEL, D_MODEL,
        nullptr, x);
}